// Circuit_71330816852913
// MI455X (gfx1250) — compile-verified
//
#include <hip/hip_runtime.h>
#include <hip/hip_bf16.h>

// MI455X / gfx1250 implementation.
//
// Stage 1 (setup_kernel, 1 block, f64 VALU):
//   H from params; U = exp(iH) via scaling-and-squaring Taylor (replaces eigh);
//   G = U^T U;  mix = G (1.25 I - G)^{-1} via complex Gauss-Jordan (G and the
//   resolvent commute);  T2 = -(k_i k_j)(0.5 I + mix).  Written to ws as f32.
//
// Stage 2 (integrate_kernel, 32 blocks x 1 wave32):
//   Each wave owns 16 independent batch rows (M=16 WMMA tile). Fixed-step RK4,
//   5 substeps per output interval (h ~ 1e-3). The complex matvec
//   A(16x64) @ T2^T is 4 chained V_WMMA_F32_16X16X4_F32 per K-slice,
//   operands streamed from LDS via ds_load_b64 in the ISA 16x4 / 4x16 layout.
//   Loop nest is K-outer so each A operand is loaded once and reused across
//   all 4 N-tiles (B operands are the only per-tile loads).
//   All state LDS-resident (56.25 KB); only output snapshots touch HBM.

#define NMODES   64
#define NM2      4096
#define BATCH    512
#define EVALPTS  200
#define SUBSTEPS 5

typedef __attribute__((ext_vector_type(2))) float v2f;
typedef __attribute__((ext_vector_type(8))) float v8f;

__device__ __forceinline__ v8f wmma4(v2f a, v2f b, v8f c) {
  // 8-arg form: (neg_a, A, neg_b, B, c_mod, C, reuse_a, reuse_b)
  return __builtin_amdgcn_wmma_f32_16x16x4_f32(false, a, false, b, (short)0, c,
                                               false, false);
}

// ---------------------------------------------------------------------------
// Setup kernel: f64, one workgroup of 256 threads. All matrices in d_ws.
// ---------------------------------------------------------------------------
__global__ __launch_bounds__(256) void setup_kernel(
    const float* __restrict__ params, const float* __restrict__ kappa,
    double* __restrict__ ws, float* __restrict__ T2f)
{
  double* HR = ws;            double* HI = ws + 1 * NM2;
  double* XR = ws + 2 * NM2;  double* XI = ws + 3 * NM2;
  double* ER = ws + 4 * NM2;  double* EI = ws + 5 * NM2;
  double* TR = ws + 6 * NM2;  double* TI = ws + 7 * NM2;
  double* PR = ws + 8 * NM2;  double* PI = ws + 9 * NM2;
  double* GR = ws + 10 * NM2; double* GI = ws + 11 * NM2;
  double* MR = ws + 12 * NM2; double* MI = ws + 13 * NM2;

  const int tid = threadIdx.x;
  __shared__ double sfr[64], sfi[64];
  __shared__ double sscale, sir, sii;
  __shared__ int sexp, spiv;

  // ---- build Hermitian H from params -------------------------------------
  for (int e = tid; e < NM2; e += 256) {
    int i = e >> 6, j = e & 63;
    double hr, hi;
    if (i < j) {
      int idx = i * (127 - i) / 2 + (j - i - 1);
      hr = (double)params[63 + idx];
      hi = (double)params[63 + 2016 + idx];
    } else if (i > j) {
      int idx = j * (127 - j) / 2 + (i - j - 1);
      hr = (double)params[63 + idx];
      hi = -(double)params[63 + 2016 + idx];
    } else {
      if (i < 63) {
        hr = (double)params[i];
      } else {
        double s = 0.0;
        for (int d = 0; d < 63; ++d) s += (double)params[d];
        hr = -s;
      }
      hi = 0.0;
    }
    HR[e] = hr;
    HI[e] = hi;
  }
  __syncthreads();

  // ---- scaling factor for exp(iH) ----------------------------------------
  if (tid == 0) {
    double mx = 0.0;
    for (int i = 0; i < 64; ++i) {
      double rs = 0.0;
      for (int j = 0; j < 64; ++j)
        rs += fabs(HR[i * 64 + j]) + fabs(HI[i * 64 + j]);
      if (rs > mx) mx = rs;
    }
    int se = 0;
    while (mx > 0.5 && se < 60) { mx *= 0.5; ++se; }
    sexp = se;
    sscale = 1.0 / (double)(1ULL << se);
  }
  __syncthreads();
  const double sc = sscale;
  const int sq = sexp;

  // X = i*H*sc ; E = I ; T = I
  for (int e = tid; e < NM2; e += 256) {
    int i = e >> 6, j = e & 63;
    XR[e] = -HI[e] * sc;
    XI[e] = HR[e] * sc;
    ER[e] = (i == j) ? 1.0 : 0.0;
    EI[e] = 0.0;
    TR[e] = ER[e];
    TI[e] = 0.0;
  }
  __syncthreads();

  // ---- Taylor series of exp(X) -------------------------------------------
  for (int term = 1; term <= 24; ++term) {
    // P = T * X
    for (int e = tid; e < NM2; e += 256) {
      int i = e >> 6, j = e & 63;
      double sr = 0.0, si = 0.0;
      for (int k = 0; k < 64; ++k) {
        double ar = TR[i * 64 + k], ai = TI[i * 64 + k];
        double br = XR[k * 64 + j], bi = XI[k * 64 + j];
        sr += ar * br - ai * bi;
        si += ar * bi + ai * br;
      }
      PR[e] = sr;
      PI[e] = si;
    }
    __syncthreads();
    double inv = 1.0 / (double)term;
    for (int e = tid; e < NM2; e += 256) {
      TR[e] = PR[e] * inv;
      TI[e] = PI[e] * inv;
      ER[e] += TR[e];
      EI[e] += TI[e];
    }
    __syncthreads();
  }

  // ---- repeated squaring: E <- E*E ---------------------------------------
  for (int q = 0; q < sq; ++q) {
    for (int e = tid; e < NM2; e += 256) {
      int i = e >> 6, j = e & 63;
      double sr = 0.0, si = 0.0;
      for (int k = 0; k < 64; ++k) {
        double ar = ER[i * 64 + k], ai = EI[i * 64 + k];
        double br = ER[k * 64 + j], bi = EI[k * 64 + j];
        sr += ar * br - ai * bi;
        si += ar * bi + ai * br;
      }
      PR[e] = sr;
      PI[e] = si;
    }
    __syncthreads();
    for (int e = tid; e < NM2; e += 256) { ER[e] = PR[e]; EI[e] = PI[e]; }
    __syncthreads();
  }

  // ---- G = U^T U (plain transpose, no conj) ------------------------------
  for (int e = tid; e < NM2; e += 256) {
    int i = e >> 6, j = e & 63;
    double sr = 0.0, si = 0.0;
    for (int k = 0; k < 64; ++k) {
      double ar = ER[k * 64 + i], ai = EI[k * 64 + i];
      double br = ER[k * 64 + j], bi = EI[k * 64 + j];
      sr += ar * br - ai * bi;
      si += ar * bi + ai * br;
    }
    GR[e] = sr;
    GI[e] = si;
  }
  __syncthreads();

  // ---- M = 1.25 I - G ; RHS P = G ----------------------------------------
  for (int e = tid; e < NM2; e += 256) {
    int i = e >> 6, j = e & 63;
    MR[e] = ((i == j) ? 1.25 : 0.0) - GR[e];
    MI[e] = -GI[e];
    PR[e] = GR[e];
    PI[e] = GI[e];
  }
  __syncthreads();

  // ---- complex Gauss-Jordan with partial pivoting: P <- M^{-1} G = mix ---
  for (int p = 0; p < 64; ++p) {
    if (tid == 0) {
      int best = p;
      double bv = -1.0;
      for (int r = p; r < 64; ++r) {
        double v = MR[r * 64 + p] * MR[r * 64 + p] +
                   MI[r * 64 + p] * MI[r * 64 + p];
        if (v > bv) { bv = v; best = r; }
      }
      spiv = best;
    }
    __syncthreads();
    int pr = spiv;
    if (pr != p) {
      for (int c = tid; c < 64; c += 256) {
        double t;
        t = MR[p * 64 + c]; MR[p * 64 + c] = MR[pr * 64 + c]; MR[pr * 64 + c] = t;
        t = MI[p * 64 + c]; MI[p * 64 + c] = MI[pr * 64 + c]; MI[pr * 64 + c] = t;
        t = PR[p * 64 + c]; PR[p * 64 + c] = PR[pr * 64 + c]; PR[pr * 64 + c] = t;
        t = PI[p * 64 + c]; PI[p * 64 + c] = PI[pr * 64 + c]; PI[pr * 64 + c] = t;
      }
    }
    __syncthreads();
    if (tid == 0) {
      double dr = MR[p * 64 + p], di = MI[p * 64 + p];
      double d2 = dr * dr + di * di;
      sir = dr / d2;
      sii = -di / d2;
    }
    __syncthreads();
    double ir = sir, ii = sii;
    for (int c = tid; c < 64; c += 256) {
      double mr = MR[p * 64 + c], mi = MI[p * 64 + c];
      MR[p * 64 + c] = mr * ir - mi * ii;
      MI[p * 64 + c] = mr * ii + mi * ir;
      double xr = PR[p * 64 + c], xi = PI[p * 64 + c];
      PR[p * 64 + c] = xr * ir - xi * ii;
      PI[p * 64 + c] = xr * ii + xi * ir;
    }
    __syncthreads();
    if (tid < 64 && tid != p) {
      sfr[tid] = MR[tid * 64 + p];
      sfi[tid] = MI[tid * 64 + p];
    }
    __syncthreads();
    for (int e = tid; e < NM2; e += 256) {
      int r = e >> 6, c = e & 63;
      if (r == p) continue;
      double fr = sfr[r], fi = sfi[r];
      double mr = MR[p * 64 + c], mi = MI[p * 64 + c];
      MR[e] -= fr * mr - fi * mi;
      MI[e] -= fr * mi + fi * mr;
      double xr = PR[p * 64 + c], xi = PI[p * 64 + c];
      PR[e] -= fr * xr - fi * xi;
      PI[e] -= fr * xi + fi * xr;
    }
    __syncthreads();
  }

  // ---- T2[n][k] = -(k_n k_k) (0.5 delta + mix), stored f32 ---------------
  for (int e = tid; e < NM2; e += 256) {
    int i = e >> 6, j = e & 63;
    double kk = fabs((double)kappa[i]) * fabs((double)kappa[j]);
    double mr = PR[e] + ((i == j) ? 0.5 : 0.0);
    T2f[e] = (float)(-kk * mr);
    T2f[NM2 + e] = (float)(-kk * PI[e]);
  }
}

// ---------------------------------------------------------------------------
// Derivative: d = i*(omega + nl2*|A|^2)*A + A @ T2^T for a 16-row tile.
// A (ytmp) and T2 live in LDS; result returned in WMMA C-layout registers.
// K-outer loop: each A re/im operand pair is loaded once per K-slice and
// reused across all 4 N-tiles; only B operands are per-tile loads.
// ---------------------------------------------------------------------------
__device__ __forceinline__ void derivative(
    const float* __restrict__ sT2r, const float* __restrict__ sT2i,
    const float* __restrict__ sAr, const float* __restrict__ sAi,
    const float* __restrict__ sOm, float nl2, v8f dR[4], v8f dI[4])
{
  const int lane = threadIdx.x & 31;
  const int mrow = lane & 15;
  const int khalf = (lane >> 4) << 1;  // lanes 16-31 carry K+2 per ISA layout
  const int mbase = (lane >> 4) << 3;  // C-layout: lanes 16-31 hold M=8..15

#pragma unroll
  for (int t = 0; t < 4; ++t) {
    dR[t] = (v8f){};
    dI[t] = (v8f){};
  }

#pragma unroll
  for (int ks = 0; ks < 16; ++ks) {
    const int k0 = ks * 4 + khalf;
    v2f Ar = *(const v2f*)(sAr + mrow * 64 + k0);
    v2f Ai = *(const v2f*)(sAi + mrow * 64 + k0);
    v2f nAi = -Ai;  // F32 WMMA has no A-negate modifier (NEG = CNeg only)
#pragma unroll
    for (int t = 0; t < 4; ++t) {
      const int nrow = t * 16 + mrow;  // T2 row = mode index
      v2f Br = *(const v2f*)(sT2r + nrow * 64 + k0);  // B[k][n] = T2[n][k]
      v2f Bi = *(const v2f*)(sT2i + nrow * 64 + k0);
      dR[t] = wmma4(Ar, Br, dR[t]);
      dR[t] = wmma4(nAi, Bi, dR[t]);
      dI[t] = wmma4(Ar, Bi, dI[t]);
      dI[t] = wmma4(Ai, Br, dI[t]);
    }
  }

  // diagonal + nonlinear term, applied in-place on the accumulators
#pragma unroll
  for (int t = 0; t < 4; ++t) {
    const int nrow = t * 16 + mrow;
    const float w0 = sOm[nrow];
#pragma unroll
    for (int r = 0; r < 8; ++r) {
      const int m = mbase + r;
      const float yr = sAr[m * 64 + nrow];
      const float yi = sAi[m * 64 + nrow];
      const float w = w0 + nl2 * (yr * yr + yi * yi);
      dR[t][r] = dR[t][r] - w * yi;
      dI[t][r] = dI[t][r] + w * yr;
    }
  }
}

// ---------------------------------------------------------------------------
// Integrator: 32 blocks x 32 threads (1 wave32 per 16-row tile). RK4.
// ---------------------------------------------------------------------------
__global__ __launch_bounds__(32) void integrate_kernel(
    const float* __restrict__ A0r, const float* __restrict__ A0i,
    const float* __restrict__ omega, const float* __restrict__ nonlin,
    const float* __restrict__ T2f, float* __restrict__ out)
{
  __shared__ __attribute__((aligned(16))) float sT2r[NM2];
  __shared__ __attribute__((aligned(16))) float sT2i[NM2];
  __shared__ __attribute__((aligned(16))) float sYr[1024], sYi[1024];
  __shared__ __attribute__((aligned(16))) float sAr[1024], sAi[1024];
  __shared__ __attribute__((aligned(16))) float sKr[1024], sKi[1024];
  __shared__ float sOm[64];

  const int lane = threadIdx.x;
  const int r0 = blockIdx.x * 16;

  for (int e = lane; e < NM2; e += 32) {
    sT2r[e] = T2f[e];
    sT2i[e] = T2f[NM2 + e];
  }
  for (int e = lane; e < 64; e += 32) sOm[e] = omega[e];
  const float nl2 = nonlin[0] * nonlin[0];

  // initial state: input modes 0..47, padding modes 48..63 = 1+0i
  for (int e = lane; e < 1024; e += 32) {
    int m = e >> 6, n = e & 63;
    float yr = 1.0f, yi = 0.0f;
    if (n < 48) {
      yr = A0r[(r0 + m) * 48 + n];
      yi = A0i[(r0 + m) * 48 + n];
    }
    sYr[e] = yr;
    sYi[e] = yi;
  }
  // snapshot t = 0
  for (int e = lane; e < 1024; e += 32) {
    int m = e >> 6, n = e & 63;
    out[(size_t)(r0 + m) * 64 + n] = sYr[e];
    out[(size_t)BATCH * 64 + (size_t)(r0 + m) * 64 + n] = sYi[e];
  }

  const float h = (1.0f / 199.0f) / (float)SUBSTEPS;
  const int l16 = lane & 15;
  const int mb = (lane >> 4) << 3;

  for (int j = 1; j < EVALPTS; ++j) {
    for (int s = 0; s < SUBSTEPS; ++s) {
      v8f dR[4], dI[4];
      for (int e = lane; e < 1024; e += 32) {
        sAr[e] = sYr[e];
        sAi[e] = sYi[e];
      }
      // k1
      derivative(sT2r, sT2i, sAr, sAi, sOm, nl2, dR, dI);
#pragma unroll
      for (int t = 0; t < 4; ++t) {
        int n = t * 16 + l16;
#pragma unroll
        for (int r = 0; r < 8; ++r) {
          int a = (mb + r) * 64 + n;
          float dr = dR[t][r], di = dI[t][r];
          sKr[a] = dr;
          sKi[a] = di;
          sAr[a] = sYr[a] + 0.5f * h * dr;
          sAi[a] = sYi[a] + 0.5f * h * di;
        }
      }
      // k2
      derivative(sT2r, sT2i, sAr, sAi, sOm, nl2, dR, dI);
#pragma unroll
      for (int t = 0; t < 4; ++t) {
        int n = t * 16 + l16;
#pragma unroll
        for (int r = 0; r < 8; ++r) {
          int a = (mb + r) * 64 + n;
          float dr = dR[t][r], di = dI[t][r];
          sKr[a] += 2.0f * dr;
          sKi[a] += 2.0f * di;
          sAr[a] = sYr[a] + 0.5f * h * dr;
          sAi[a] = sYi[a] + 0.5f * h * di;
        }
      }
      // k3
      derivative(sT2r, sT2i, sAr, sAi, sOm, nl2, dR, dI);
#pragma unroll
      for (int t = 0; t < 4; ++t) {
        int n = t * 16 + l16;
#pragma unroll
        for (int r = 0; r < 8; ++r) {
          int a = (mb + r) * 64 + n;
          float dr = dR[t][r], di = dI[t][r];
          sKr[a] += 2.0f * dr;
          sKi[a] += 2.0f * di;
          sAr[a] = sYr[a] + h * dr;
          sAi[a] = sYi[a] + h * di;
        }
      }
      // k4 + state update
      derivative(sT2r, sT2i, sAr, sAi, sOm, nl2, dR, dI);
      const float h6 = h / 6.0f;
#pragma unroll
      for (int t = 0; t < 4; ++t) {
        int n = t * 16 + l16;
#pragma unroll
        for (int r = 0; r < 8; ++r) {
          int a = (mb + r) * 64 + n;
          sYr[a] += h6 * (sKr[a] + dR[t][r]);
          sYi[a] += h6 * (sKi[a] + dI[t][r]);
        }
      }
    }
    // snapshot j
    for (int e = lane; e < 1024; e += 32) {
      int m = e >> 6, n = e & 63;
      size_t base = (size_t)j * 2 * BATCH * 64;
      out[base + (size_t)(r0 + m) * 64 + n] = sYr[e];
      out[base + (size_t)BATCH * 64 + (size_t)(r0 + m) * 64 + n] = sYi[e];
    }
  }
}

// ---------------------------------------------------------------------------
extern "C" void kernel_launch(void* const* d_in, const int* in_sizes, int n_in,
                              void* d_out, int out_size, void* d_ws,
                              size_t ws_size, hipStream_t stream)
{
  const float* A0r = (const float*)d_in[0];
  const float* A0i = (const float*)d_in[1];
  const float* omega = (const float*)d_in[2];
  const float* kappa = (const float*)d_in[3];
  const float* nonlin = (const float*)d_in[4];
  const float* params = (const float*)d_in[5];

  double* ws = (double*)d_ws;
  float* T2f = (float*)((char*)d_ws + (size_t)14 * NM2 * sizeof(double));

  setup_kernel<<<1, 256, 0, stream>>>(params, kappa, ws, T2f);
  integrate_kernel<<<32, 32, 0, stream>>>(A0r, A0i, omega, nonlin, T2f,
                                          (float*)d_out);
}